// LinearCDE_43542378447334
// MI455X (gfx1250) — compile-verified
//
#include <hip/hip_runtime.h>

typedef float v2f __attribute__((ext_vector_type(2)));
typedef float v8f __attribute__((ext_vector_type(8)));

#define BB    8
#define SS    512
#define DATA  63
#define HID   512
#define OMEGA 64
#define NWG   64      // scan workgroups; each owns HID/NWG = 8 h-rows (1 per wave)

// ---------------- ws layout (floats) ----------------
// C  : [SS][BB][OMEGA]            = 262144
// U  : [HID][OMEGA]               = 32768
// Y0 : [BB][HID]                  = 4096   (ping)
// Y1 : [BB][HID]                  = 4096   (pong)
// bar: 2 x unsigned
#define WS_C   0
#define WS_U   (WS_C + SS * BB * OMEGA)
#define WS_Y0  (WS_U + HID * OMEGA)
#define WS_Y1  (WS_Y0 + BB * HID)
#define WS_BAR (WS_Y1 + BB * HID)

__global__ __launch_bounds__(256) void lcde_setup(
    const float* __restrict__ X, const float* __restrict__ init_matrix,
    const float* __restrict__ init_bias, const float* __restrict__ b_A,
    const float* __restrict__ vf_B, float* __restrict__ ws,
    float* __restrict__ out) {
  int idx = blockIdx.x * 256 + threadIdx.x;
  // control path C[s][b][o]; o==0 -> t_s = s/(S-1), else X[b,s,o-1]
  if (idx < SS * BB * OMEGA) {
    int s = idx / (BB * OMEGA);
    int r = idx - s * (BB * OMEGA);
    int b = r / OMEGA;
    int o = r - b * OMEGA;
    float v = (o == 0) ? ((float)s * (1.0f / (float)(SS - 1)))
                       : X[((size_t)b * SS + s) * DATA + (o - 1)];
    ws[WS_C + idx] = v;
  }
  // U = b_A (reshaped [HID][OMEGA]) + vf_B
  if (idx < HID * OMEGA) ws[WS_U + idx] = b_A[idx] + vf_B[idx];
  // y0[b][h] = X[b,0,:] . init_matrix[h,:] + init_bias[h]
  if (idx < BB * HID) {
    int b = idx >> 9;
    int h = idx & (HID - 1);
    float acc = init_bias[h];
    const float* xr = X + (size_t)b * SS * DATA;  // s = 0
    const float* mr = init_matrix + (size_t)h * DATA;
#pragma unroll 9
    for (int k = 0; k < DATA; ++k) acc += xr[k] * mr[k];
    ws[WS_Y0 + idx] = acc;
    out[(size_t)b * SS * HID + h] = acc;  // ys[b, 0, h]
  }
  if (idx == 0) {
    unsigned* bar = (unsigned*)(ws + WS_BAR);
    bar[0] = 0u;
    bar[1] = 0u;
  }
}

__device__ __forceinline__ void grid_barrier(unsigned* bar) {
  __threadfence();  // make this WG's global writes visible at device scope
  __syncthreads();
  if (threadIdx.x == 0) {
    unsigned g = __hip_atomic_load(&bar[1], __ATOMIC_ACQUIRE, __HIP_MEMORY_SCOPE_AGENT);
    unsigned v = __hip_atomic_fetch_add(&bar[0], 1u, __ATOMIC_ACQ_REL, __HIP_MEMORY_SCOPE_AGENT);
    if (v == NWG - 1) {
      __hip_atomic_store(&bar[0], 0u, __ATOMIC_RELAXED, __HIP_MEMORY_SCOPE_AGENT);
      __hip_atomic_fetch_add(&bar[1], 1u, __ATOMIC_RELEASE, __HIP_MEMORY_SCOPE_AGENT);
    } else {
      while (__hip_atomic_load(&bar[1], __ATOMIC_ACQUIRE, __HIP_MEMORY_SCOPE_AGENT) == g) {
        __builtin_amdgcn_s_sleep(2);
      }
    }
  }
  __syncthreads();
}

// Load one 8-K chunk (A fragments from LDS, B fragments from W_A in L2)
#define LOADCH(buf, kk)                                        \
  do {                                                         \
    Af[buf][0] = *(const v2f*)(Ys + (kk));                     \
    Af[buf][1] = *(const v2f*)(Ys + (kk) + 4);                 \
    Bf[buf][0][0] = *(const v2f*)(Wp0 + (kk));                 \
    Bf[buf][0][1] = *(const v2f*)(Wp0 + (kk) + 4);             \
    Bf[buf][1][0] = *(const v2f*)(Wp1 + (kk));                 \
    Bf[buf][1][1] = *(const v2f*)(Wp1 + (kk) + 4);             \
    Bf[buf][2][0] = *(const v2f*)(Wp2 + (kk));                 \
    Bf[buf][2][1] = *(const v2f*)(Wp2 + (kk) + 4);             \
    Bf[buf][3][0] = *(const v2f*)(Wp3 + (kk));                 \
    Bf[buf][3][1] = *(const v2f*)(Wp3 + (kk) + 4);             \
  } while (0)

// Issue the 8 WMMAs of one chunk (independent accs back-to-back)
#define WMMA1(buf, j, acc, t)                                               \
  acc = __builtin_amdgcn_wmma_f32_16x16x4_f32(false, Af[buf][j], false,     \
                                              Bf[buf][t][j], (short)0, acc, \
                                              false, false)
#define WMMACH(buf)                                            \
  do {                                                         \
    WMMA1(buf, 0, acc0, 0);                                    \
    WMMA1(buf, 0, acc1, 1);                                    \
    WMMA1(buf, 0, acc2, 2);                                    \
    WMMA1(buf, 0, acc3, 3);                                    \
    WMMA1(buf, 1, acc0, 0);                                    \
    WMMA1(buf, 1, acc1, 1);                                    \
    WMMA1(buf, 1, acc2, 2);                                    \
    WMMA1(buf, 1, acc3, 3);                                    \
  } while (0)

// Persistent scan: 64 WGs x 256 threads (8 waves). Wave w of WG g owns
// h = g*8 + w, i.e. N-columns [h*64, h*64+64) of the per-step
// [16 x 32768] = Ypad[16x512] @ W_A^T[512x32768] GEMM (8 real batch rows,
// 8 zero-pad rows). K = 512 via v_wmma_f32_16x16x4_f32, software-pipelined
// so ~2 chunks of loads stay in flight while WMMAs issue.
__global__ __launch_bounds__(256, 1) void lcde_scan(
    const float* __restrict__ W,   // W_A [32768][512] row-major (row n = h*64+o)
    float* __restrict__ ws, float* __restrict__ out) {
  __shared__ float Ylds[16 * 516];   // padded row stride 516 -> conflict-free
  __shared__ float Clds[BB * OMEGA]; // C_s staged per step

  const float* Cg = ws + WS_C;
  const float* U  = ws + WS_U;
  float* Ybuf0 = ws + WS_Y0;
  float* Ybuf1 = ws + WS_Y1;
  unsigned* bar = (unsigned*)(ws + WS_BAR);

  const int tid   = threadIdx.x;
  const int wave  = tid >> 5;
  const int lane  = tid & 31;
  const int h     = blockIdx.x * 8 + wave;  // 0..511
  const int m     = lane & 15;              // M (A) or N (B, C/D) index
  const int khalf = (lane >> 4) << 1;       // lanes 0-15: K=+0,+1; 16-31: K=+2,+3

  // Per-wave constants: U[h][o] for this wave's 64 o-columns, W row pointers.
  float Ureg[4];
#pragma unroll
  for (int t = 0; t < 4; ++t) Ureg[t] = U[h * OMEGA + t * 16 + m];
  const float* Wp0 = W + (size_t)(h * OMEGA + 0 * 16 + m) * HID + khalf;
  const float* Wp1 = W + (size_t)(h * OMEGA + 1 * 16 + m) * HID + khalf;
  const float* Wp2 = W + (size_t)(h * OMEGA + 2 * 16 + m) * HID + khalf;
  const float* Wp3 = W + (size_t)(h * OMEGA + 3 * 16 + m) * HID + khalf;

  // Zero the pad rows (M = 8..15) of Ylds once.
  for (int i = tid; i < 8 * 516; i += 256) Ylds[8 * 516 + i] = 0.0f;

  const float dt = 1.0f / (float)SS;

  for (int s = 1; s < SS; ++s) {
    const float* Yr = (s & 1) ? Ybuf0 : Ybuf1;  // read  (s-1)&1
    float*       Yw = (s & 1) ? Ybuf1 : Ybuf0;  // write s&1
    __syncthreads();  // previous iteration done reading LDS
    for (int i = tid; i < BB * HID; i += 256)
      Ylds[(i >> 9) * 516 + (i & (HID - 1))] = Yr[i];
    for (int i = tid; i < BB * OMEGA; i += 256)
      Clds[i] = Cg[(size_t)s * (BB * OMEGA) + i];
    __syncthreads();

    v8f acc0 = {}, acc1 = {}, acc2 = {}, acc3 = {};
    const float* Ys = &Ylds[m * 516 + khalf];

    v2f Af[2][2];
    v2f Bf[2][4][2];
    LOADCH(0, 0);
    LOADCH(1, 8);
    // steady state: consume chunk k (buf k/8 mod 2), immediately refill that
    // buffer with chunk k+16 -> loads stay ~2 chunks ahead of their use.
#pragma unroll 2
    for (int k = 0; k < HID - 16; k += 8) {
      const int cur = (k >> 3) & 1;
      WMMACH(cur);
      LOADCH(cur, k + 16);
    }
    WMMACH(0);  // chunk k=496 lives in buf 0
    WMMACH(1);  // chunk k=504 lives in buf 1

    // Epilogue: upd[b] = sum_o (vfA[b,o] + U[h,o]) * C_s[b,o]
    float p[BB] = {0, 0, 0, 0, 0, 0, 0, 0};
    if (lane < 16) {
      v8f accs[4] = {acc0, acc1, acc2, acc3};
#pragma unroll
      for (int t = 0; t < 4; ++t) {
        int o = t * 16 + lane;
#pragma unroll
        for (int r = 0; r < BB; ++r)
          p[r] += (accs[t][r] + Ureg[t]) * Clds[r * OMEGA + o];
      }
    }
#pragma unroll
    for (int mask = 1; mask < 16; mask <<= 1) {
#pragma unroll
      for (int r = 0; r < BB; ++r) p[r] += __shfl_xor(p[r], mask, 32);
    }
    if (lane < BB) {
      float ynew = Ylds[lane * 516 + h] + p[lane] * dt;
      Yw[lane * HID + h] = ynew;
      out[(size_t)lane * SS * HID + (size_t)s * HID + h] = ynew;
    }
    grid_barrier(bar);
  }
}

extern "C" void kernel_launch(void* const* d_in, const int* in_sizes, int n_in,
                              void* d_out, int out_size, void* d_ws, size_t ws_size,
                              hipStream_t stream) {
  const float* X           = (const float*)d_in[0];
  const float* init_matrix = (const float*)d_in[1];
  const float* init_bias   = (const float*)d_in[2];
  const float* W_A         = (const float*)d_in[3];
  const float* b_A         = (const float*)d_in[4];
  const float* vf_B        = (const float*)d_in[5];
  float* out = (float*)d_out;
  float* ws  = (float*)d_ws;

  lcde_setup<<<dim3((SS * BB * OMEGA + 255) / 256), dim3(256), 0, stream>>>(
      X, init_matrix, init_bias, b_A, vf_B, ws, out);
  lcde_scan<<<dim3(NWG), dim3(256), 0, stream>>>(W_A, ws, out);
}